// EEG_STDPModel_10823317586346
// MI455X (gfx1250) — compile-verified
//
#include <hip/hip_runtime.h>
#include <hip/hip_bf16.h>

// ---------------------------------------------------------------------------
// SNN liquid-state machine, restructured:
//   Phase 1: per-batch-lane LIF recurrence over T=1000 steps, 20-dim state
//            per layer; layer-3 folded into a 20-dim linear accumulator `a`
//            (v3_t = W3 @ a_t by linearity since layer 3 never spikes).
//   Phase 2: out = exp(A[4096x20] @ W3^T[20x500]) via V_WMMA_F32_16X16X4_F32.
// ---------------------------------------------------------------------------

#define B_TOTAL   4096
#define T_STEPS   1000
#define TT        4        // time-chunk (weight-broadcast amortization)
#define VTH       5.0f

typedef float v2f __attribute__((ext_vector_type(2)));
typedef float v8f __attribute__((ext_vector_type(8)));

// ------------------------------ Phase 1 ------------------------------------
// Only 128 waves exist in the whole grid (4096 lanes), so occupancy is
// irrelevant; (128, 1) lets the compiler take the full VGPR budget and avoid
// spilling the TT=4 chunk-body working set (~280 live floats).
__global__ __launch_bounds__(128, 1) void snn_recurrence(
    const float* __restrict__ x,    // [4096][14][1000]
    const float* __restrict__ W1,   // [20][14]
    const float* __restrict__ W2,   // [20][20]
    float* __restrict__ a_out)      // [4096][20]
{
  // Transposed, pre-scaled (by 1/tau = 0.5) weights in LDS.
  __shared__ __align__(16) float sW1[14 * 20];  // sW1[l*20+i] = 0.5*W1[i][l]
  __shared__ __align__(16) float sW2[20 * 20];  // sW2[j*20+i] = 0.5*W2[i][j]

  for (int idx = threadIdx.x; idx < 14 * 20; idx += blockDim.x) {
    int l = idx / 20, i = idx % 20;
    sW1[l * 20 + i] = 0.5f * W1[i * 14 + l];
  }
  for (int idx = threadIdx.x; idx < 20 * 20; idx += blockDim.x) {
    int j = idx / 20, i = idx % 20;
    sW2[j * 20 + i] = 0.5f * W2[i * 20 + j];
  }
  __syncthreads();

  const int b = blockIdx.x * blockDim.x + threadIdx.x;   // one lane per batch
  const float* xb = x + (size_t)b * 14 * T_STEPS;

  float v1[20], v2[20], aa[20];
#pragma unroll
  for (int i = 0; i < 20; ++i) { v1[i] = 0.f; v2[i] = 0.f; aa[i] = 0.f; }

  for (int tc = 0; tc < T_STEPS; tc += TT) {
    // ---- layer 1: h1/2 for TT steps, weight-broadcast amortized ----
    float acc[20][TT];
#pragma unroll
    for (int i = 0; i < 20; ++i)
#pragma unroll
      for (int t = 0; t < TT; ++t) acc[i][t] = 0.f;

#pragma unroll
    for (int l = 0; l < 14; ++l) {
      float4 xv = *(const float4*)(xb + l * T_STEPS + tc);  // 16B aligned
      float xt[TT] = {xv.x, xv.y, xv.z, xv.w};
      const float4* wrow = (const float4*)(sW1 + l * 20);   // 80B aligned
#pragma unroll
      for (int i4 = 0; i4 < 5; ++i4) {
        float4 w = wrow[i4];
        float wv[4] = {w.x, w.y, w.z, w.w};
#pragma unroll
        for (int q = 0; q < 4; ++q)
#pragma unroll
          for (int t = 0; t < TT; ++t)
            acc[i4 * 4 + q][t] = __builtin_fmaf(wv[q], xt[t], acc[i4 * 4 + q][t]);
      }
    }

    // ---- LIF layer 1 (sequential in t); spikes overwrite acc in place ----
#pragma unroll
    for (int t = 0; t < TT; ++t)
#pragma unroll
      for (int i = 0; i < 20; ++i) {
        float v = __builtin_fmaf(v1[i], 0.5f, acc[i][t]);   // v/2 + h/2
        bool sp = (v >= VTH);
        acc[i][t] = sp ? 1.0f : 0.0f;                       // s1
        v1[i] = sp ? 0.0f : v;                              // hard reset
      }

    // ---- layer 2: h2/2 from binary spikes, weight-broadcast amortized ----
    float hcc[20][TT];
#pragma unroll
    for (int i = 0; i < 20; ++i)
#pragma unroll
      for (int t = 0; t < TT; ++t) hcc[i][t] = 0.f;

#pragma unroll
    for (int j = 0; j < 20; ++j) {
      const float4* wrow = (const float4*)(sW2 + j * 20);
#pragma unroll
      for (int i4 = 0; i4 < 5; ++i4) {
        float4 w = wrow[i4];
        float wv[4] = {w.x, w.y, w.z, w.w};
#pragma unroll
        for (int q = 0; q < 4; ++q)
#pragma unroll
          for (int t = 0; t < TT; ++t)
            hcc[i4 * 4 + q][t] =
                __builtin_fmaf(wv[q], acc[j][t], hcc[i4 * 4 + q][t]);
      }
    }

    // ---- LIF layer 2 + linear layer-3 accumulator a = (a + s2)/2 ----
#pragma unroll
    for (int t = 0; t < TT; ++t)
#pragma unroll
      for (int i = 0; i < 20; ++i) {
        float v = __builtin_fmaf(v2[i], 0.5f, hcc[i][t]);
        bool sp = (v >= VTH);
        v2[i] = sp ? 0.0f : v;
        aa[i] = __builtin_fmaf(aa[i], 0.5f, sp ? 0.5f : 0.0f);
      }
  }

  float* ao = a_out + (size_t)b * 20;
#pragma unroll
  for (int i4 = 0; i4 < 5; ++i4) {
    float4 s = make_float4(aa[i4 * 4 + 0], aa[i4 * 4 + 1],
                           aa[i4 * 4 + 2], aa[i4 * 4 + 3]);
    *(float4*)(ao + i4 * 4) = s;   // 80B row -> 16B aligned
  }
}

// ------------------------------ Phase 2 ------------------------------------
// out[b][n] = exp( sum_j a[b][j] * W3[n][j] ),  b tile 16 x n tile 16, K=20.
// One wave (32 threads) per tile; V_WMMA_F32_16X16X4_F32, 5 K-chunks.
__global__ __launch_bounds__(32) void snn_readout_wmma(
    const float* __restrict__ a,    // [4096][20]
    const float* __restrict__ W3,   // [500][20]
    float* __restrict__ out)        // [4096][500]
{
  const int lane = threadIdx.x;        // 0..31
  const int half = lane >> 4;          // 0 | 1
  const int lr   = lane & 15;
  const int b0   = blockIdx.x * 16;    // 256 batch tiles
  const int n0   = blockIdx.y * 16;    // 32 neuron tiles (512 >= 500)

  int nClamped = n0 + lr;              // clamp loads for the partial tile
  if (nClamped > 499) nClamped = 499;

  // A 16x4 f32 layout: lane -> row = lane&15, VGPR{0,1} = K {2*half, 2*half+1}
  const float* arow = a  + (size_t)(b0 + lr) * 20 + 2 * half;
  // B 4x16 f32 layout: lane -> col = lane&15, VGPR{0,1} = K {2*half, 2*half+1}
  const float* wrow = W3 + (size_t)nClamped * 20 + 2 * half;

  v8f c = {0.f, 0.f, 0.f, 0.f, 0.f, 0.f, 0.f, 0.f};
#pragma unroll
  for (int k = 0; k < 5; ++k) {                 // K = 20 = 5 x 4
    v2f av = *(const v2f*)(arow + 4 * k);       // 8B aligned (even offsets)
    v2f bv = *(const v2f*)(wrow + 4 * k);
    c = __builtin_amdgcn_wmma_f32_16x16x4_f32(
        /*neg_a=*/false, av, /*neg_b=*/false, bv,
        /*c_mod=*/(short)0, c, /*reuse_a=*/false, /*reuse_b=*/false);
  }

  const int n = n0 + lr;                        // D: n = lane&15
#pragma unroll
  for (int r = 0; r < 8; ++r) {                 // D: m = r + 8*half
    const int m = r + 8 * half;
    if (n < 500) out[(size_t)(b0 + m) * 500 + n] = expf(c[r]);
  }
}

// ------------------------------ Launch -------------------------------------
extern "C" void kernel_launch(void* const* d_in, const int* in_sizes, int n_in,
                              void* d_out, int out_size, void* d_ws,
                              size_t ws_size, hipStream_t stream) {
  const float* x  = (const float*)d_in[0];   // [4096,14,1000]
  const float* W1 = (const float*)d_in[1];   // [20,14]
  const float* W2 = (const float*)d_in[2];   // [20,20]
  const float* W3 = (const float*)d_in[3];   // [500,20]
  float* out  = (float*)d_out;               // [4096,500]
  float* a_ws = (float*)d_ws;                // [4096,20] scratch (320 KB)

  snn_recurrence<<<dim3(B_TOTAL / 128), dim3(128), 0, stream>>>(x, W1, W2, a_ws);
  snn_readout_wmma<<<dim3(B_TOTAL / 16, 32), dim3(32), 0, stream>>>(a_ws, W3, out);
}